// S4_PTD_Kernel_49778670960670
// MI455X (gfx1250) — compile-verified
//
#include <hip/hip_runtime.h>
#include <math.h>

#define HH   512
#define NH   32
#define LL   8192
// l = 16*a + b ; a in [0,512), b in [0,16)
// per h: real GEMM M=512, N=16, K=64  ->  32 m-tiles x 16 k-steps of v_wmma_f32_16x16x4_f32
// one wave = one (h, m-tile); 8 waves/block; grid = 512*32/8 = 2048 blocks

typedef __attribute__((ext_vector_type(2))) float v2f;
typedef __attribute__((ext_vector_type(8))) float v8f;

__device__ __forceinline__ float red2pi(float y) {
    // cheap range reduction for hardware sin/cos at large |y|
    const float INV2PI = 0.15915494309189535f;
    const float TWOPI  = 6.283185307179586f;
    return fmaf(-TWOPI, rintf(y * INV2PI), y);
}

__global__ __launch_bounds__(256) void s4d_wmma_kernel(
    const float* __restrict__ log_dt,      // (H)
    const float* __restrict__ log_A_real,  // (H,NH)
    const float* __restrict__ A_imag,      // (H,NH)
    const float* __restrict__ Bp,          // (H,NH,2)
    const float* __restrict__ Cp,          // (H,NH,2)
    float* __restrict__ out)               // (H,L)
{
    const int lane  = threadIdx.x & 31;
    const int wave  = threadIdx.x >> 5;
    const int gtile = blockIdx.x * 8 + wave;   // 0 .. 16383
    const int h     = gtile >> 5;              // 32 tiles per h
    const int mtile = gtile & 31;              // 16-row block of 'a'
    const int n     = lane;                    // this lane's state index (Nh == 32 == wave32)

    // ---- per-(h,n) parameters, one complex state per lane ----
    const int  hn   = h * NH + n;
    float dt   = __expf(log_dt[h]);
    float Are  = -__expf(log_A_real[hn]);      // A = -exp(log_A_real) - i*A_imag
    float Aim  = -A_imag[hn];
    float dtAr = Are * dt;
    float dtAi = Aim * dt;
    float Br = Bp[2*hn+0], Bi = Bp[2*hn+1];
    float Cr = Cp[2*hn+0], Ci = Cp[2*hn+1];
    // E = exp(dtA) - 1
    float e0 = __expf(dtAr);
    float s0, c0;  __sincosf(dtAi, &s0, &c0);
    float Er = e0 * c0 - 1.0f;
    float Ei = e0 * s0;
    // BC = B*C
    float BCr = Br*Cr - Bi*Ci;
    float BCi = Br*Ci + Bi*Cr;
    // num = BC * E
    float Nr = BCr*Er - BCi*Ei;
    float Ni = BCr*Ei + BCi*Er;
    // Ceff = num / A = num * conj(A) / |A|^2
    float inv = 1.0f / (Are*Are + Aim*Aim);
    float Cer = (Nr*Are + Ni*Aim) * inv;
    float Cei = (Ni*Are - Nr*Aim) * inv;

    // ---- fragment geometry ----
    const int   b    = lane & 15;              // N column (and power for B-frag)
    const int   kh   = lane >> 4;              // which K-pair this lane holds
    const int   arow = lane & 15;              // M row within tile
    const float pA   = 16.0f * (float)(mtile * 16 + arow);   // exponent multiplier for A-frag
    const float pB   = (float)b;                              // exponent multiplier for B-frag

    v8f acc = {};
    #pragma unroll
    for (int kk = 0; kk < 16; ++kk) {
        const int np = 2*kk + kh;              // state index n' feeding this lane's K-pair
        float dr = __shfl(dtAr, np, 32);
        float di = __shfl(dtAi, np, 32);
        float cr = __shfl(Cer,  np, 32);
        float ci = __shfl(Cei,  np, 32);

        // B fragment: V = r^b = exp(dr*b) * cis(di*b)   -> (Re, Im)
        float eb = __expf(dr * pB);
        float sb, cb;  __sincosf(red2pi(di * pB), &sb, &cb);
        v2f bf;
        bf.x = eb * cb;
        bf.y = eb * sb;

        // A fragment: U = Ceff * r^(16a)                -> (Re, -Im)
        float ea = __expf(dr * pA);
        float sa, ca;  __sincosf(red2pi(di * pA), &sa, &ca);
        float wr = ea * ca, wi = ea * sa;
        v2f af;
        af.x =  cr*wr - ci*wi;
        af.y = -(cr*wi + ci*wr);

        // D(16x16) += A(16x4) x B(4x16), f32 exact
        acc = __builtin_amdgcn_wmma_f32_16x16x4_f32(
            /*neg_a=*/false, af, /*neg_b=*/false, bf,
            /*c_mod=*/(short)0, acc, /*reuse_a=*/false, /*reuse_b=*/false);
    }

    // ---- writeback: D VGPR j -> M = j + 8*kh, N = b ----
    float* op = out + (size_t)h * LL + (size_t)(mtile * 16) * 16;
    #pragma unroll
    for (int j = 0; j < 8; ++j) {
        int M = j + 8*kh;
        op[M*16 + b] = 2.0f * acc[j];
    }
}

extern "C" void kernel_launch(void* const* d_in, const int* in_sizes, int n_in,
                              void* d_out, int out_size, void* d_ws, size_t ws_size,
                              hipStream_t stream) {
    const float* log_dt     = (const float*)d_in[0];
    const float* log_A_real = (const float*)d_in[1];
    const float* A_imag     = (const float*)d_in[2];
    const float* Bp         = (const float*)d_in[3];
    const float* Cp         = (const float*)d_in[4];
    // d_in[5] is L (int scalar) == 8192, baked in as LL
    float* out = (float*)d_out;

    dim3 grid(HH * 32 / 8);   // 2048 blocks, 8 waves (one 16x16 tile each) per block
    dim3 block(256);
    s4d_wmma_kernel<<<grid, block, 0, stream>>>(log_dt, log_A_real, A_imag, Bp, Cp, out);
}